// SoftAndLayer_9672266350668
// MI455X (gfx1250) — compile-verified
//
#include <hip/hip_runtime.h>

// Problem constants (from reference): x (B,D), w (L,D), out (B,L)
constexpr int kB = 1024;
constexpr int kL = 512;
constexpr int kD = 512;

constexpr int kBT = 32;                 // B-tile per block
constexpr int kLT = 64;                 // L-tile per block
constexpr int kDK = 32;                 // D-chunk staged in LDS
constexpr int kNC = kD / kDK;           // 16 chunks
constexpr int kStride = 36;             // LDS row stride in floats (144B: 16B-aligned, conflict-free)

typedef float float4v __attribute__((ext_vector_type(4)));
typedef int   int4v   __attribute__((ext_vector_type(4)));

typedef __attribute__((address_space(1))) int4v* g_int4_ptr;
typedef __attribute__((address_space(3))) int4v* l_int4_ptr;

// ---- CDNA5 async global->LDS (ASYNCcnt) with inline-asm fallback ----------
__device__ __forceinline__ void async_ld_b128(const float* g, float* lds) {
#if __has_builtin(__builtin_amdgcn_global_load_async_to_lds_b128)
  __builtin_amdgcn_global_load_async_to_lds_b128(
      (g_int4_ptr)(unsigned long long)g,   // global src (AS1 int4*)
      (l_int4_ptr)lds,                     // LDS dst (AS3 int4*)
      /*offset=*/0, /*cpol=*/0);
#else
  unsigned loff =
      (unsigned)(unsigned long long)(__attribute__((address_space(3))) char*)lds;
  asm volatile("global_load_async_to_lds_b128 %0, %1, off"
               :: "v"(loff), "v"(g) : "memory");
#endif
}

#if __has_builtin(__builtin_amdgcn_s_wait_asynccnt)
#define WAIT_ASYNC(n) __builtin_amdgcn_s_wait_asynccnt(n)
#else
#define WAIT_ASYNC(n) asm volatile("s_wait_asynccnt %0" :: "i"(n) : "memory")
#endif

// ---------------------------------------------------------------------------
__global__ __launch_bounds__(256)
void tropical_minmax_kernel(const float* __restrict__ x,
                            const float* __restrict__ w,
                            float* __restrict__ out) {
  __shared__ float xs[2][kBT][kStride];   // 32 x-rows per buffer
  __shared__ float ws[2][kLT][kStride];   // 64 w-rows per buffer

  const int tid = threadIdx.x;
  const int tx = tid & 15;   // l-lane (0..15) -> l = ll + tx + 16j, j=0..3
  const int ty = tid >> 4;   // b-lane (0..15) -> b = bb + ty + 16i, i=0..1
  const int bb = blockIdx.y * kBT;
  const int ll = blockIdx.x * kLT;

  // Tile staging: x tile = 32x32 floats = 256 float4 (1/thread),
  //               w tile = 64x32 floats = 512 float4 (2/thread).
  const int r0 = tid >> 3;         // 0..31
  const int c0 = (tid & 7) * 4;    // column in floats (0..28)

  const float* xg0 = x + (bb + r0) * kD + c0;          // x rows 0..31 of tile
  const float* wg0 = w + (ll + r0) * kD + c0;          // w rows 0..31
  const float* wg1 = w + (ll + 32 + r0) * kD + c0;     // w rows 32..63

  float acc[2][4];
#pragma unroll
  for (int i = 0; i < 2; ++i)
#pragma unroll
    for (int j = 0; j < 4; ++j) acc[i][j] = __builtin_inff();

  // ---- prologue: stage chunk 0 -------------------------------------------
  async_ld_b128(xg0, &xs[0][r0][c0]);
  float4v wv0 = *(const float4v*)wg0;
  float4v wv1 = *(const float4v*)wg1;

  for (int c = 0; c < kNC; ++c) {
    const int buf = c & 1;

    // Transform w -> 1 - clip(w,0,1) once per tile, store to LDS.
    float4v t0, t1;
#pragma unroll
    for (int e = 0; e < 4; ++e) {
      t0[e] = 1.0f - fminf(fmaxf(wv0[e], 0.0f), 1.0f);
      t1[e] = 1.0f - fminf(fmaxf(wv1[e], 0.0f), 1.0f);
    }
    *(float4v*)&ws[buf][r0][c0]      = t0;
    *(float4v*)&ws[buf][32 + r0][c0] = t1;

    // Prefetch next chunk into the other buffer (async pipe + regular loads).
    if (c + 1 < kNC) {
      const int dd = (c + 1) * kDK;
      async_ld_b128(xg0 + dd, &xs[buf ^ 1][r0][c0]);
      wv0 = *(const float4v*)(wg0 + dd);
      wv1 = *(const float4v*)(wg1 + dd);
      WAIT_ASYNC(1);   // chunk c's async load retired (in-order), c+1 in flight
    } else {
      WAIT_ASYNC(0);
    }
    __syncthreads();   // also drains DScnt for the ws stores

    // ---- compute: 2x4 register tile, 4 k-steps per iteration -------------
#pragma unroll
    for (int k4 = 0; k4 < kDK / 4; ++k4) {
      float4v xv[2], vv[4];
#pragma unroll
      for (int i = 0; i < 2; ++i)
        xv[i] = *(const float4v*)&xs[buf][ty + 16 * i][k4 * 4];
#pragma unroll
      for (int j = 0; j < 4; ++j)
        vv[j] = *(const float4v*)&ws[buf][tx + 16 * j][k4 * 4];
#pragma unroll
      for (int i = 0; i < 2; ++i)
#pragma unroll
        for (int j = 0; j < 4; ++j)
#pragma unroll
          for (int e = 0; e < 4; ++e)
            acc[i][j] = fminf(acc[i][j], fmaxf(xv[i][e], vv[j][e]));
    }
    __syncthreads();   // buffer reuse fence
  }

  // ---- epilogue: scatter the 2x4 tile ------------------------------------
#pragma unroll
  for (int i = 0; i < 2; ++i) {
    const int b = bb + ty + 16 * i;
#pragma unroll
    for (int j = 0; j < 4; ++j) {
      const int l = ll + tx + 16 * j;
      out[b * kL + l] = acc[i][j];
    }
  }
}

// ---------------------------------------------------------------------------
extern "C" void kernel_launch(void* const* d_in, const int* in_sizes, int n_in,
                              void* d_out, int out_size, void* d_ws, size_t ws_size,
                              hipStream_t stream) {
  (void)in_sizes; (void)n_in; (void)out_size; (void)d_ws; (void)ws_size;
  const float* x = (const float*)d_in[0];     // (1024, 512)
  const float* w = (const float*)d_in[1];     // (512, 512)
  float* out = (float*)d_out;                 // (1024, 512)

  dim3 grid(kL / kLT, kB / kBT);              // (8, 32) = 256 blocks
  tropical_minmax_kernel<<<grid, 256, 0, stream>>>(x, w, out);
}